// GraphSAGENodeBaseline_1065151889634
// MI455X (gfx1250) — compile-verified
//
#include <hip/hip_runtime.h>

#define HID 64

typedef __attribute__((ext_vector_type(2))) float v2f;
typedef __attribute__((ext_vector_type(8))) float v8f;

// D = A(16x4) * B(4x16) + C, full fp32 via CDNA5 WMMA.
__device__ __forceinline__ v8f wmma_k4(v2f a, v2f b, v8f c) {
  return __builtin_amdgcn_wmma_f32_16x16x4_f32(
      /*neg_a=*/false, a, /*neg_b=*/false, b,
      /*c_mod=*/(short)0, c, /*reuse_a=*/false, /*reuse_b=*/false);
}

// Guaranteed hardware fp32 atomic add: non-returning (STOREcnt, fire and
// forget), device scope so the RMW is performed in the shared 192MB L2.
__device__ __forceinline__ void atomic_add_f32_hw(float* addr, float val) {
  asm volatile("global_atomic_add_f32 %0, %1, off scope:SCOPE_DEV"
               :
               : "v"(addr), "v"(val)
               : "memory");
}

__global__ void zero_f32(float* __restrict__ p, long long n) {
  long long i = (long long)blockIdx.x * blockDim.x + threadIdx.x;
  if (i < n) p[i] = 0.0f;
}

__global__ void degree_kernel(const int* __restrict__ dst, float* __restrict__ deg, int E) {
  int e = blockIdx.x * blockDim.x + threadIdx.x;
  if (e < E) atomic_add_f32_hw(&deg[dst[e]], 1.0f);
}

// One (edge, channel) pair per thread: coalesced gather of h[src], coalesced
// hardware f32 atomic scatter into agg[dst]. agg + h are L2-resident (51MB each).
__global__ void scatter_kernel(const float* __restrict__ h, const int* __restrict__ src,
                               const int* __restrict__ dst, float* __restrict__ agg,
                               long long total) {
  long long t = (long long)blockIdx.x * blockDim.x + threadIdx.x;
  if (t >= total) return;
  int e = (int)(t >> 6);
  int c = (int)(t & 63);
  int s = src[e];
  int d = dst[e];
  atomic_add_f32_hw(&agg[(long long)d * HID + c], h[(long long)s * HID + c]);
}

// h = relu(x @ W_proj + b), x:[n,165], W:[165,64]. 32 rows/block, 4 waves.
// Each wave owns one 16-col tile and BOTH 16-row tiles: every B fragment is
// loaded from LDS once and used by two WMMAs (2 accumulators), cutting LDS
// traffic from 4 to 3 dwords per WMMA. K padded 165 -> 168 (zeros).
// LDS strides padded (172 / 68) so fragment reads don't bank-alias.
// 67.7KB LDS -> dynamic allocation (WGP has 320KB).
#define PROJ_K 168
#define PROJ_SX_STRIDE 172
#define PROJ_SW_STRIDE 68
#define PROJ_LDS_FLOATS (32 * PROJ_SX_STRIDE + PROJ_K * PROJ_SW_STRIDE)

__global__ __launch_bounds__(128) void proj_kernel(const float* __restrict__ x,
                                                   const float* __restrict__ W,
                                                   const float* __restrict__ bias,
                                                   float* __restrict__ h, int n) {
  extern __shared__ float smem[];
  float(*sX)[PROJ_SX_STRIDE] = (float(*)[PROJ_SX_STRIDE])smem;
  float(*sW)[PROJ_SW_STRIDE] = (float(*)[PROJ_SW_STRIDE])(smem + 32 * PROJ_SX_STRIDE);

  const int tid = threadIdx.x;
  const int rowBase = blockIdx.x * 32;

  for (int i = tid; i < 32 * PROJ_K; i += 128) {
    int r = i / PROJ_K, c = i % PROJ_K;
    int row = rowBase + r;
    sX[r][c] = (row < n && c < 165) ? x[(long long)row * 165 + c] : 0.0f;
  }
  for (int i = tid; i < PROJ_K * 64; i += 128) {
    int r = i >> 6, c = i & 63;
    sW[r][c] = (r < 165) ? W[r * 64 + c] : 0.0f;
  }
  __syncthreads();

  const int wave = tid >> 5;
  const int lane = tid & 31;
  const int hf = lane >> 4;  // which 16-lane half
  const int m = lane & 15;
  const int n0 = wave * 16;

  v8f acc0 = {};
  v8f acc1 = {};
  for (int k = 0; k < PROJ_K; k += 4) {
    const int ka = k + 2 * hf;
    v2f b, a0, a1;
    b[0] = sW[ka][n0 + m];
    b[1] = sW[ka + 1][n0 + m];
    a0[0] = sX[m][ka];
    a0[1] = sX[m][ka + 1];
    a1[0] = sX[16 + m][ka];
    a1[1] = sX[16 + m][ka + 1];
    acc0 = wmma_k4(a0, b, acc0);
    acc1 = wmma_k4(a1, b, acc1);
  }

  const float bv = bias[n0 + m];
  for (int v = 0; v < 8; ++v) {
    int r0 = rowBase + v + 8 * hf;
    int r1 = r0 + 16;
    if (r0 < n) {
      float val = acc0[v] + bv;
      h[(long long)r0 * HID + n0 + m] = val > 0.0f ? val : 0.0f;
    }
    if (r1 < n) {
      float val = acc1[v] + bv;
      h[(long long)r1 * HID + n0 + m] = val > 0.0f ? val : 0.0f;
    }
  }
}

// h_out = relu((agg/max(deg,1)) @ Wl + bl + h_in @ Wr). 32 rows/block,
// 4 waves; wave = one col-tile x two row-tiles with shared B fragments;
// both GEMMs accumulate into the same C fragments.
__global__ __launch_bounds__(128) void combine_kernel(
    const float* __restrict__ agg, const float* __restrict__ deg,
    const float* __restrict__ h_in, const float* __restrict__ Wl,
    const float* __restrict__ bl, const float* __restrict__ Wr,
    float* __restrict__ h_out, int n) {
  __shared__ float sAa[32][68];
  __shared__ float sAh[32][68];
  __shared__ float sWl[64][68];
  __shared__ float sWr[64][68];
  const int tid = threadIdx.x;
  const int rowBase = blockIdx.x * 32;

  for (int i = tid; i < 32 * 64; i += 128) {
    int r = i >> 6, c = i & 63;
    int row = rowBase + r;
    float va = 0.0f, vh = 0.0f;
    if (row < n) {
      float d = deg[row];
      d = d > 1.0f ? d : 1.0f;
      va = agg[(long long)row * HID + c] / d;
      vh = h_in[(long long)row * HID + c];
    }
    sAa[r][c] = va;
    sAh[r][c] = vh;
  }
  for (int i = tid; i < 64 * 64; i += 128) {
    int r = i >> 6, c = i & 63;
    sWl[r][c] = Wl[i];
    sWr[r][c] = Wr[i];
  }
  __syncthreads();

  const int wave = tid >> 5;
  const int lane = tid & 31;
  const int hf = lane >> 4;
  const int m = lane & 15;
  const int n0 = wave * 16;

  v8f acc0 = {};
  v8f acc1 = {};
  for (int k = 0; k < 64; k += 4) {
    const int ka = k + 2 * hf;
    v2f b, a0, a1;
    b[0] = sWl[ka][n0 + m];
    b[1] = sWl[ka + 1][n0 + m];
    a0[0] = sAa[m][ka];
    a0[1] = sAa[m][ka + 1];
    a1[0] = sAa[16 + m][ka];
    a1[1] = sAa[16 + m][ka + 1];
    acc0 = wmma_k4(a0, b, acc0);
    acc1 = wmma_k4(a1, b, acc1);
  }
  for (int k = 0; k < 64; k += 4) {
    const int ka = k + 2 * hf;
    v2f b, a0, a1;
    b[0] = sWr[ka][n0 + m];
    b[1] = sWr[ka + 1][n0 + m];
    a0[0] = sAh[m][ka];
    a0[1] = sAh[m][ka + 1];
    a1[0] = sAh[16 + m][ka];
    a1[1] = sAh[16 + m][ka + 1];
    acc0 = wmma_k4(a0, b, acc0);
    acc1 = wmma_k4(a1, b, acc1);
  }

  const float bv = bl[n0 + m];
  for (int v = 0; v < 8; ++v) {
    int r0 = rowBase + v + 8 * hf;
    int r1 = r0 + 16;
    if (r0 < n) {
      float val = acc0[v] + bv;
      h_out[(long long)r0 * HID + n0 + m] = val > 0.0f ? val : 0.0f;
    }
    if (r1 < n) {
      float val = acc1[v] + bv;
      h_out[(long long)r1 * HID + n0 + m] = val > 0.0f ? val : 0.0f;
    }
  }
}

__global__ void cls_kernel(const float* __restrict__ h, const float* __restrict__ Wc,
                           const float* __restrict__ bc, float* __restrict__ out, int n) {
  int i = blockIdx.x * blockDim.x + threadIdx.x;
  if (i >= n) return;
  const float4* hp = (const float4*)(h + (long long)i * HID);
  const float4* wp = (const float4*)Wc;
  float acc = bc[0];
#pragma unroll
  for (int j = 0; j < HID / 4; ++j) {
    float4 hv = hp[j];
    float4 wv = wp[j];
    acc += hv.x * wv.x + hv.y * wv.y + hv.z * wv.z + hv.w * wv.w;
  }
  out[i] = acc;
}

extern "C" void kernel_launch(void* const* d_in, const int* in_sizes, int n_in,
                              void* d_out, int out_size, void* d_ws, size_t ws_size,
                              hipStream_t stream) {
  const float* x   = (const float*)d_in[0];
  const int* eidx  = (const int*)d_in[1];
  const float* Wp  = (const float*)d_in[2];
  const float* bp  = (const float*)d_in[3];
  const float* W1l = (const float*)d_in[4];
  const float* b1l = (const float*)d_in[5];
  const float* W1r = (const float*)d_in[6];
  const float* W2l = (const float*)d_in[7];
  const float* b2l = (const float*)d_in[8];
  const float* W2r = (const float*)d_in[9];
  const float* Wc  = (const float*)d_in[10];
  const float* bc  = (const float*)d_in[11];

  const int N = in_sizes[0] / 165;
  const int E = in_sizes[1] / 2;
  const int* src = eidx;
  const int* dst = eidx + E;

  float* A   = (float*)d_ws;            // h0 / h2     [N*64]
  float* B   = A + (size_t)N * HID;     // h1          [N*64]
  float* AGG = B + (size_t)N * HID;     // scatter acc [N*64]
  float* DEG = AGG + (size_t)N * HID;   // degrees     [N]

  const long long nh = (long long)N * HID;
  const long long te = (long long)E * HID;
  const unsigned gz_nh = (unsigned)((nh + 255) / 256);
  const unsigned gz_n  = (unsigned)((N + 255) / 256);
  const unsigned g_sc  = (unsigned)((te + 255) / 256);
  const unsigned g_row = (unsigned)((N + 31) / 32);
  const size_t proj_lds = (size_t)PROJ_LDS_FLOATS * sizeof(float);

  // degrees (shared by both convs) + first aggregation buffer
  zero_f32<<<gz_nh, 256, 0, stream>>>(AGG, nh);
  zero_f32<<<gz_n, 256, 0, stream>>>(DEG, (long long)N);
  degree_kernel<<<(E + 255) / 256, 256, 0, stream>>>(dst, DEG, E);

  // h0 = relu(x @ W_proj + b_proj)
  proj_kernel<<<g_row, 128, proj_lds, stream>>>(x, Wp, bp, A, N);

  // conv 1: A -> B
  scatter_kernel<<<g_sc, 256, 0, stream>>>(A, src, dst, AGG, te);
  combine_kernel<<<g_row, 128, 0, stream>>>(AGG, DEG, A, W1l, b1l, W1r, B, N);

  // conv 2: B -> A
  zero_f32<<<gz_nh, 256, 0, stream>>>(AGG, nh);
  scatter_kernel<<<g_sc, 256, 0, stream>>>(B, src, dst, AGG, te);
  combine_kernel<<<g_row, 128, 0, stream>>>(AGG, DEG, B, W2l, b2l, W2r, A, N);

  // out = h2 @ W_cls + b_cls
  cls_kernel<<<(N + 255) / 256, 256, 0, stream>>>(A, Wc, bc, (float*)d_out, N);
}